// CustomBERTModel_29996051595344
// MI455X (gfx1250) — compile-verified
//
#include <hip/hip_runtime.h>
#include <hip/hip_bf16.h>

#define MASK_ID 103
#define TOPK    20
#define NB      16
#define NS      256
#define NV      30522

typedef __attribute__((ext_vector_type(2))) float v2f;
typedef __attribute__((ext_vector_type(4))) float v4f;
typedef __attribute__((ext_vector_type(8))) float v8f;

// ---------------------------------------------------------------------------
// 1) Zero the 500MB output with wide non-temporal stores (bandwidth floor).
// ---------------------------------------------------------------------------
__global__ __launch_bounds__(256) void k_zero(v4f* __restrict__ out, long n4) {
    long i = (long)blockIdx.x * 256 + threadIdx.x;
    if (i < n4) {
        v4f z = {0.f, 0.f, 0.f, 0.f};
        __builtin_nontemporal_store(z, out + i);
    }
}

// ---------------------------------------------------------------------------
// 2) Find the [MASK] position per row (first occurrence, exactly one exists).
// ---------------------------------------------------------------------------
__global__ __launch_bounds__(256) void k_maskpos(const int* __restrict__ ids,
                                                 int* __restrict__ pos) {
    __shared__ int smin;
    int b = blockIdx.x;
    if (threadIdx.x == 0) smin = NS;
    __syncthreads();
    if (ids[b * NS + threadIdx.x] == MASK_ID) atomicMin(&smin, (int)threadIdx.x);
    __syncthreads();
    if (threadIdx.x == 0) pos[b] = (smin == NS) ? 0 : smin;  // argmax semantics
}

// ---------------------------------------------------------------------------
// 3) Top-20 of the mask row (reads only 16 x 30522 floats, not full logits).
//    Per-thread private top-20, then 20 rounds of block argmax over LDS.
// ---------------------------------------------------------------------------
__global__ __launch_bounds__(256) void k_topk(const float* __restrict__ logits,
                                              const int* __restrict__ pos,
                                              float* __restrict__ topVals,
                                              int* __restrict__ topIdx) {
    const int b   = blockIdx.x;
    const int tid = threadIdx.x;
    const float* row = logits + ((long)b * NS + pos[b]) * (long)NV;

    float pv[TOPK];
    int   pi[TOPK];
#pragma unroll
    for (int k = 0; k < TOPK; ++k) { pv[k] = -__builtin_inff(); pi[k] = -1; }

    for (int i = tid; i < NV; i += 256) {
        float v = row[i];
        if (v > pv[TOPK - 1]) {                 // rejected with 1 compare usually
            int p = TOPK - 1;
            while (p > 0 && pv[p - 1] < v) { pv[p] = pv[p - 1]; pi[p] = pi[p - 1]; --p; }
            pv[p] = v; pi[p] = i;
        }
    }

    __shared__ float cv[256 * TOPK];
    __shared__ int   ci[256 * TOPK];
#pragma unroll
    for (int k = 0; k < TOPK; ++k) { cv[tid * TOPK + k] = pv[k]; ci[tid * TOPK + k] = pi[k]; }
    __syncthreads();

    __shared__ float rv[256];
    __shared__ int   rp[256];
    for (int k = 0; k < TOPK; ++k) {
        float best = -__builtin_inff();
        int   bpos = 0;
        for (int t = tid; t < 256 * TOPK; t += 256)
            if (cv[t] > best) { best = cv[t]; bpos = t; }
        rv[tid] = best; rp[tid] = bpos;
        __syncthreads();
        for (int off = 128; off > 0; off >>= 1) {
            if (tid < off && rv[tid + off] > rv[tid]) {
                rv[tid] = rv[tid + off]; rp[tid] = rp[tid + off];
            }
            __syncthreads();
        }
        if (tid == 0) {
            topVals[b * TOPK + k] = rv[0];
            topIdx [b * TOPK + k] = ci[rp[0]];
            cv[rp[0]] = -__builtin_inff();
        }
        __syncthreads();
    }
}

// ---------------------------------------------------------------------------
// 4) out_vals = vals @ W^T + b via V_WMMA_F32_16X16X4_F32 (M=16,K=20,N=20),
//    softmax over 20, scatter 320 probs into the zeroed output.
//    Single wave32 block; EXEC all-1s at the WMMAs (branchless padding).
// ---------------------------------------------------------------------------
__global__ __launch_bounds__(32) void k_agent(const float* __restrict__ topVals,
                                              const int* __restrict__ topIdx,
                                              const float* __restrict__ W,
                                              const float* __restrict__ bias,
                                              const int* __restrict__ pos,
                                              float* __restrict__ out) {
    const int lane = threadIdx.x;            // 0..31, wave32
    const int hi   = (lane >= 16) ? 1 : 0;   // which K-half this lane holds
    const int lm   = lane & 15;              // M (A) / N (B,C,D) row-in-tile

    __shared__ float ov[NB][TOPK];           // out_vals, then probs

#if __has_builtin(__builtin_amdgcn_wmma_f32_16x16x4_f32)
#pragma unroll
    for (int nt = 0; nt < 2; ++nt) {         // N tiles: cols 0..15 and 16..19(+pad)
        const int   n     = lm + nt * 16;
        const int   nclmp = (n < TOPK) ? n : (TOPK - 1);
        const float nmask = (n < TOPK) ? 1.f : 0.f;

        v8f c;
        const float bval = bias[nclmp] * nmask;  // preload bias into accumulator
#pragma unroll
        for (int r = 0; r < 8; ++r) c[r] = bval;

#pragma unroll
        for (int kk = 0; kk < 5; ++kk) {     // K = 20 = 5 x 4
            const int kb = kk * 4 + hi * 2;  // A layout: lanes0-15 K=0,1; 16-31 K=2,3
            v2f a, bm;
            a.x  = topVals[lm * TOPK + kb];
            a.y  = topVals[lm * TOPK + kb + 1];
            bm.x = W[nclmp * TOPK + kb]     * nmask;   // B[k][n] = W[n][k]
            bm.y = W[nclmp * TOPK + kb + 1] * nmask;
            c = __builtin_amdgcn_wmma_f32_16x16x4_f32(
                    false, a, false, bm, (short)0, c, false, false);
        }
        // C/D layout: VGPR r -> M=r (lanes0-15) / M=r+8 (lanes16-31), N=lane%16
#pragma unroll
        for (int r = 0; r < 8; ++r) {
            const int m = r + hi * 8;
            if (n < TOPK) ov[m][n] = c[r];
        }
    }
#else
    // Fallback (should not trigger per probe: builtin is declared with 8 args).
    for (int t = lane; t < NB * TOPK; t += 32) {
        const int m = t / TOPK, n = t % TOPK;
        float acc = bias[n];
        for (int k = 0; k < TOPK; ++k) acc += topVals[m * TOPK + k] * W[n * TOPK + k];
        ov[m][n] = acc;
    }
#endif
    __syncthreads();

    // Softmax per row: lanes 0..15 each own one row of 20.
    if (lane < NB) {
        float mx = ov[lane][0];
#pragma unroll
        for (int k = 1; k < TOPK; ++k) mx = fmaxf(mx, ov[lane][k]);
        float s = 0.f;
#pragma unroll
        for (int k = 0; k < TOPK; ++k) { float e = expf(ov[lane][k] - mx); ov[lane][k] = e; s += e; }
        const float inv = 1.f / s;
#pragma unroll
        for (int k = 0; k < TOPK; ++k) ov[lane][k] *= inv;
    }
    __syncthreads();

    // Scatter 320 probabilities into the zeroed output.
    for (int t = lane; t < NB * TOPK; t += 32) {
        const int b = t / TOPK, k = t % TOPK;
        const long off = ((long)b * NS + pos[b]) * (long)NV + topIdx[b * TOPK + k];
        out[off] = ov[b][k];
    }
}

// ---------------------------------------------------------------------------
extern "C" void kernel_launch(void* const* d_in, const int* in_sizes, int n_in,
                              void* d_out, int out_size, void* d_ws, size_t ws_size,
                              hipStream_t stream) {
    const float* logits = (const float*)d_in[0];   // [16,256,30522] f32
    const int*   ids    = (const int*)  d_in[1];   // [16,256] i32
    const float* W      = (const float*)d_in[2];   // [20,20] f32
    const float* bias   = (const float*)d_in[3];   // [20] f32
    float*       out    = (float*)d_out;

    // workspace layout (tiny): pos[16] @0, topVals[320] @256, topIdx[320] @2048
    int*   pos     = (int*)  d_ws;
    float* topVals = (float*)((char*)d_ws + 256);
    int*   topIdx  = (int*)  ((char*)d_ws + 2048);

    const long n4 = (long)out_size / 4;            // 31,254,528 float4 stores
    const int  zb = (int)((n4 + 255) / 256);

    k_zero   <<<zb, 256, 0, stream>>>((v4f*)out, n4);
    k_maskpos<<<NB, 256, 0, stream>>>(ids, pos);
    k_topk   <<<NB, 256, 0, stream>>>(logits, pos, topVals, topIdx);
    k_agent  <<<1,   32, 0, stream>>>(topVals, topIdx, W, bias, pos, out);
}